// MoERouter_84954453115199
// MI455X (gfx1250) — compile-verified
//
#include <hip/hip_runtime.h>

typedef __attribute__((ext_vector_type(2))) float v2f;
typedef __attribute__((ext_vector_type(8))) float v8f;

#define H        2048
#define HP       2052      // padded LDS row stride (floats): bank = (4m+k)%64, conflict-free
#define E        64
#define TILE_M   16
#define NTHREADS 128       // 4 waves (wave32)

__global__ __launch_bounds__(NTHREADS)
void moe_router_kernel(const float* __restrict__ hidden,
                       const float* __restrict__ lnw,
                       const float* __restrict__ lnb,
                       const float* __restrict__ gw,
                       float* __restrict__ out,
                       int N)
{
    extern __shared__ float smem[];
    float* Xl = smem;                   // 16 x 2052 f32 normalized activations
    float* Lg = smem + TILE_M * HP;     // 16 x 64  f32 clipped logits

    const int t       = threadIdx.x;
    const int lane    = t & 31;
    const int wave    = t >> 5;
    const int rowbase = blockIdx.x * TILE_M;

    // ---- stage 16x2048 f32 tile into LDS (coalesced float4) ----
    {
        const float4* src = (const float4*)(hidden + (size_t)rowbase * H);
        for (int i = 0; i < (TILE_M * H / 4) / NTHREADS; ++i) {
            int f  = i * NTHREADS + t;       // float4 index within tile
            int r  = f >> 9;                 // / (H/4 == 512)
            int c4 = f & 511;
            float4 v = src[(size_t)r * (H / 4) + c4];
            *(float4*)(&Xl[r * HP + c4 * 4]) = v;
        }
    }
    __syncthreads();

    // ---- LayerNorm: 8 threads per row, shuffle-reduce, normalize in place ----
    {
        int r = t >> 3;                      // row 0..15
        int p = t & 7;                       // chunk 0..7 (256 elems each)
        float s = 0.f, sq = 0.f;
        float* xr = &Xl[r * HP + p * (H / 8)];
        for (int i = 0; i < H / 8; ++i) { float v = xr[i]; s += v; sq += v * v; }
        s += __shfl_xor(s, 1);  sq += __shfl_xor(sq, 1);
        s += __shfl_xor(s, 2);  sq += __shfl_xor(sq, 2);
        s += __shfl_xor(s, 4);  sq += __shfl_xor(sq, 4);
        float mu   = s * (1.0f / H);
        float var  = sq * (1.0f / H) - mu * mu;
        float rstd = rsqrtf(var + 1e-5f);
        const float* w = lnw + p * (H / 8);
        const float* b = lnb + p * (H / 8);
        for (int i = 0; i < H / 8; ++i) {
            float v = (xr[i] - mu) * rstd * w[i] + b[i];
            v = (v != v) ? 0.0f : v;                // nan -> 0
            v = fminf(fmaxf(v, -50.0f), 50.0f);     // safe_clamp (covers +-inf)
            xr[i] = v;
        }
    }
    __syncthreads();

    // ---- GEMM via V_WMMA_F32_16X16X4_F32: wave w -> experts [16w, 16w+16) ----
    {
        const int   ntbase = wave * 16;
        const int   m      = lane & 15;
        const int   kofs   = (lane < 16) ? 0 : 2;     // ISA 16x4 f32 A/B K split
        const float* arow  = &Xl[m * HP + kofs];
        const float* grow  = gw + (size_t)(ntbase + m) * H + kofs;  // B[k][n]=gw[n][k]
        v8f acc = {};
#pragma unroll 4
        for (int k0 = 0; k0 < H; k0 += 4) {
            v2f a, bb;
            a.x  = arow[k0];     a.y  = arow[k0 + 1];   // ds_load_b64
            bb.x = grow[k0];     bb.y = grow[k0 + 1];   // global_load_b64, L0-reused
            acc = __builtin_amdgcn_wmma_f32_16x16x4_f32(
                      false, a, false, bb, (short)0, acc, false, false);
        }
        // clip logits to [-10,10], scatter D (C-layout) into LDS logits tile
#pragma unroll
        for (int i = 0; i < 8; ++i) {
            float v  = fminf(fmaxf(acc[i], -10.0f), 10.0f);
            int   mm = (lane < 16) ? i : (i + 8);
            Lg[mm * E + ntbase + m] = v;
        }
    }
    __syncthreads();

    // ---- softmax + top-2 per row (threads 0..15), clip probs to [1e-4, 1] ----
    if (t < TILE_M) {
        int row = rowbase + t;
        const float* lg = &Lg[t * E];
        float mx = -3.0e38f;
        for (int e = 0; e < E; ++e) mx = fmaxf(mx, lg[e]);
        float ssum = 0.f;
        for (int e = 0; e < E; ++e) ssum += __expf(lg[e] - mx);
        float inv = 1.0f / ssum;
        float p1 = -1.f, p2 = -1.f; int i1 = 0, i2 = 0;
        for (int e = 0; e < E; ++e) {
            float pe = __expf(lg[e] - mx) * inv;
            pe = fminf(fmaxf(pe, 1e-4f), 1.0f);
            if (pe > p1)      { p2 = p1; i2 = i1; p1 = pe; i1 = e; }
            else if (pe > p2) { p2 = pe; i2 = e; }
        }
        float ps = fmaxf(p1 + p2, 1e-4f);
        out[row * 2 + 0] = p1 / ps;
        out[row * 2 + 1] = p2 / ps;
        float* oidx = out + (size_t)N * 2;
        oidx[row * 2 + 0] = (float)i1;
        oidx[row * 2 + 1] = (float)i2;
    }

    // ---- coalesced logits writeback ----
    {
        float* olog = out + (size_t)N * 4 + (size_t)rowbase * E;
        for (int j = t; j < TILE_M * E; j += NTHREADS) olog[j] = Lg[j];
    }
}

extern "C" void kernel_launch(void* const* d_in, const int* in_sizes, int n_in,
                              void* d_out, int out_size, void* d_ws, size_t ws_size,
                              hipStream_t stream) {
    const float* hidden = (const float*)d_in[0];
    const float* lnw    = (const float*)d_in[1];
    const float* lnb    = (const float*)d_in[2];
    const float* gw     = (const float*)d_in[3];
    float* out = (float*)d_out;
    int N = in_sizes[0] / H;                        // 16384 tokens
    size_t smem = (size_t)(TILE_M * HP + TILE_M * E) * sizeof(float);
    moe_router_kernel<<<N / TILE_M, NTHREADS, smem, stream>>>(hidden, lnw, lnb, gw, out, N);
}